// Critic_506806140988
// MI455X (gfx1250) — compile-verified
//
#include <hip/hip_runtime.h>
#include <math.h>

typedef __attribute__((ext_vector_type(2))) float v2f;
typedef __attribute__((ext_vector_type(8))) float v8f;

#define H 256
#define NPTS 1000
#define BATCH 256

// ---------------------------------------------------------------------------
// tanh: native gfx1250 v_tanh_f32 when available, else exp-based (exact limits)
// ---------------------------------------------------------------------------
__device__ __forceinline__ float ftanh(float x) {
#if __has_builtin(__builtin_amdgcn_tanhf)
    return __builtin_amdgcn_tanhf(x);
#else
    float e = __expf(2.0f * x);                 // v_exp_f32 path
    return 1.0f - __fdividef(2.0f, e + 1.0f);
#endif
}

// ---------------------------------------------------------------------------
// Prep: W2_L = ref_w_L @ enc_w  ([H,2]), bias2_L = ref_w_L@enc_b + ref_b_L,
//       tb_L = bias2_L + q_b_L.   grid(3) x block(256)
// ---------------------------------------------------------------------------
struct PrepArgs {
    const float* ref_w[3];
    const float* ref_b[3];
    const float* q_b[3];
    const float* enc_w;   // [H,2] row-major
    const float* enc_b;   // [H]
    float* w2;            // [3][2][H]
    float* bias2;         // [3][H]
    float* tb;            // [3][H]
};

__global__ __launch_bounds__(H) void critic_prep(PrepArgs p) {
    const int L = blockIdx.x;
    const int h = threadIdx.x;
    const float* rw = p.ref_w[L] + h * H;
    float a0 = 0.f, a1 = 0.f, ab = 0.f;
#pragma unroll 4
    for (int k = 0; k < H; ++k) {
        float r = rw[k];
        a0 = fmaf(r, p.enc_w[k * 2 + 0], a0);
        a1 = fmaf(r, p.enc_w[k * 2 + 1], a1);
        ab = fmaf(r, p.enc_b[k], ab);
    }
    float b2 = ab + p.ref_b[L][h];
    p.w2[L * 2 * H + h]     = a0;
    p.w2[L * 2 * H + H + h] = a1;
    p.bias2[L * H + h]      = b2;
    p.tb[L * H + h]         = b2 + p.q_b[L][h];
}

__global__ __launch_bounds__(256) void critic_zero(float* p) {
    p[blockIdx.x * 256 + threadIdx.x] = 0.0f;
}

// ---------------------------------------------------------------------------
// WMMA fp32 GEMM: C[256,256] = A[256,256] @ W^T + bias (+optional relu)
//   C[m,n] = sum_k A[m,k] * W[n,k] + bias[n]
// One 16x16 tile per wave via V_WMMA_F32_16X16X4_F32 (K stepped by 4).
// grid(32) x block(256)  -> 256 waves == 256 tiles, EXEC all ones.
// ---------------------------------------------------------------------------
__global__ __launch_bounds__(256) void critic_gemm256_wmma(
    const float* __restrict__ A, const float* __restrict__ W,
    const float* __restrict__ bias, float* __restrict__ C, int doRelu)
{
    const int lane = threadIdx.x & 31;
    const int wave = threadIdx.x >> 5;
    const int tile = blockIdx.x * 8 + wave;   // 0..255
    const int mt = tile >> 4;
    const int nt = tile & 15;
    const int half = lane >> 4;               // 0 or 1
    const int r    = lane & 15;

    // A-frag: lane r holds A[mt*16+r, k+2*half .. k+2*half+1]
    const float* aRow = A + (mt * 16 + r) * H + 2 * half;
    // B-frag (K x N = W^T): lane r holds W[nt*16+r, k+2*half .. +1]
    const float* bRow = W + (nt * 16 + r) * H + 2 * half;

    v8f c = {};
#pragma unroll 4
    for (int k = 0; k < H; k += 4) {
        v2f a = *(const v2f*)(aRow + k);
        v2f b = *(const v2f*)(bRow + k);
        c = __builtin_amdgcn_wmma_f32_16x16x4_f32(
                /*neg_a=*/false, a, /*neg_b=*/false, b,
                /*c_mod=*/(short)0, c, /*reuse_a=*/false, /*reuse_b=*/false);
    }

    const int col = nt * 16 + r;
    const float bv = bias[col];
#pragma unroll
    for (int j = 0; j < 8; ++j) {
        int row = mt * 16 + j + 8 * half;
        float v = c[j] + bv;
        if (doRelu) v = fmaxf(v, 0.0f);
        C[row * H + col] = v;
    }
}

// ---------------------------------------------------------------------------
// Fused attention step (rank-2 collapsed):
//   e[h,n] = w0[h]*x0[n] + w1[h]*x1[n] + bias2[h]
//   u[n]   = sum_h v[h]*tanh(e[h,n] + q[b,h])          (t = q + bias2 folded)
//   probs  = softmax_n(u);  c = sum_n probs*(x0,x1)
//   hy[b,h]= w0[h]*c0 + w1[h]*c1 + bias2[h]
// grid(B) x block(256).  h-loop outermost: one ds_load_b128 broadcast feeds
// 4 independent fma+tanh chains (LDS reuse + TRANS latency hiding).
// ---------------------------------------------------------------------------
__global__ __launch_bounds__(H) void critic_attend(
    const float* __restrict__ staticf,   // flat [B, 2*N] view
    const float* __restrict__ w2L,       // [2][H]
    const float* __restrict__ bias2L,    // [H]
    const float* __restrict__ vL,        // [H]
    const float* __restrict__ tmat,      // [B,H]  (q + bias2)
    float* __restrict__ hy)              // [B,H] out
{
    __shared__ float4 wvt[H];            // {w0, w1, v, t}
    __shared__ float4 sred[H];

    const int tid = threadIdx.x;
    const int b   = blockIdx.x;

    wvt[tid] = make_float4(w2L[tid], w2L[H + tid], vL[tid], tmat[b * H + tid]);
    __syncthreads();

    // Gather this thread's 4 columns (pad with 0 -> finite garbage, masked later)
    float x0[4], x1[4], acc[4];
#pragma unroll
    for (int it = 0; it < 4; ++it) {
        const int n = tid + it * H;
        const bool ok = (n < NPTS);
        x0[it] = ok ? staticf[b * 2 * NPTS + n] : 0.0f;
        x1[it] = ok ? staticf[b * 2 * NPTS + NPTS + n] : 0.0f;
        acc[it] = 0.0f;
    }

#pragma unroll 2
    for (int h = 0; h < H; ++h) {
        float4 w = wvt[h];               // ds_load_b128, broadcast across wave
#pragma unroll
        for (int it = 0; it < 4; ++it) {
            float z = fmaf(w.x, x0[it], fmaf(w.y, x1[it], w.w));
            acc[it] = fmaf(w.z, ftanh(z), acc[it]);
        }
    }

    float u[4];
    float lmax = -__builtin_inff();
#pragma unroll
    for (int it = 0; it < 4; ++it) {
        const int n = tid + it * H;
        u[it] = (n < NPTS) ? acc[it] : -__builtin_inff();
        lmax = fmaxf(lmax, u[it]);
    }

    // block max of u
    sred[tid].x = lmax;
    __syncthreads();
    for (int s = 128; s > 0; s >>= 1) {
        if (tid < s) sred[tid].x = fmaxf(sred[tid].x, sred[tid + s].x);
        __syncthreads();
    }
    const float m = sred[0].x;
    __syncthreads();

    // partial sums: exp-sum, exp*x0, exp*x1
    float ps = 0.f, pe0 = 0.f, pe1 = 0.f;
#pragma unroll
    for (int it = 0; it < 4; ++it) {
        float p = __expf(u[it] - m);     // exp(-inf) == 0 for padded lanes
        ps += p;
        pe0 = fmaf(p, x0[it], pe0);
        pe1 = fmaf(p, x1[it], pe1);
    }
    sred[tid] = make_float4(ps, pe0, pe1, 0.f);
    __syncthreads();
    for (int s = 128; s > 0; s >>= 1) {
        if (tid < s) {
            float4 a = sred[tid], c = sred[tid + s];
            a.x += c.x; a.y += c.y; a.z += c.z;
            sred[tid] = a;
        }
        __syncthreads();
    }
    const float inv = __fdividef(1.0f, sred[0].x);
    const float c0 = sred[0].y * inv;
    const float c1 = sred[0].z * inv;

    const float4 w = wvt[tid];
    hy[b * H + tid] = fmaf(w.x, c0, fmaf(w.y, c1, bias2L[tid]));
}

// ---------------------------------------------------------------------------
// fc2: out[b] = dot(reluOut[b,:], fc2_w) + fc2_b[0].  grid(B) x block(256)
// ---------------------------------------------------------------------------
__global__ __launch_bounds__(H) void critic_fc2(
    const float* __restrict__ r, const float* __restrict__ w,
    const float* __restrict__ bias, float* __restrict__ out)
{
    __shared__ float sred[H];
    const int b = blockIdx.x, tid = threadIdx.x;
    sred[tid] = r[b * H + tid] * w[tid];
    __syncthreads();
    for (int s = 128; s > 0; s >>= 1) {
        if (tid < s) sred[tid] += sred[tid + s];
        __syncthreads();
    }
    if (tid == 0) out[b] = sred[0] + bias[0];
}

// ---------------------------------------------------------------------------
extern "C" void kernel_launch(void* const* d_in, const int* in_sizes, int n_in,
                              void* d_out, int out_size, void* d_ws, size_t ws_size,
                              hipStream_t stream) {
    (void)in_sizes; (void)n_in; (void)out_size; (void)ws_size;

    const float* staticf = (const float*)d_in[0];
    const float* enc_w   = (const float*)d_in[1];
    const float* enc_b   = (const float*)d_in[2];
    const float* ref_w[3] = {(const float*)d_in[3],  (const float*)d_in[8],  (const float*)d_in[13]};
    const float* ref_b[3] = {(const float*)d_in[4],  (const float*)d_in[9],  (const float*)d_in[14]};
    const float* q_w[3]   = {(const float*)d_in[5],  (const float*)d_in[10], (const float*)d_in[15]};
    const float* q_b[3]   = {(const float*)d_in[6],  (const float*)d_in[11], (const float*)d_in[16]};
    const float* vv[3]    = {(const float*)d_in[7],  (const float*)d_in[12], (const float*)d_in[17]};
    const float* fc1_w = (const float*)d_in[18];
    const float* fc1_b = (const float*)d_in[19];
    const float* fc2_w = (const float*)d_in[20];
    const float* fc2_b = (const float*)d_in[21];

    float* ws    = (float*)d_ws;
    float* w2    = ws;                       // 3*2*256 = 1536
    float* bias2 = ws + 1536;                // 768
    float* tb    = ws + 2304;                // 768
    float* hy    = ws + 3072;                // 65536
    float* t     = ws + 3072 + 65536;        // 65536 (reused as relu out)

    PrepArgs pa;
    for (int L = 0; L < 3; ++L) { pa.ref_w[L] = ref_w[L]; pa.ref_b[L] = ref_b[L]; pa.q_b[L] = q_b[L]; }
    pa.enc_w = enc_w; pa.enc_b = enc_b;
    pa.w2 = w2; pa.bias2 = bias2; pa.tb = tb;

    critic_prep<<<3, H, 0, stream>>>(pa);
    critic_zero<<<BATCH, 256, 0, stream>>>(hy);

    for (int L = 0; L < 3; ++L) {
        // t = hy @ q_w^T + (q_b + bias2)
        critic_gemm256_wmma<<<32, 256, 0, stream>>>(hy, q_w[L], tb + L * H, t, 0);
        // hy = attend(coords, W2_L, bias2_L, v_L, t)
        critic_attend<<<BATCH, H, 0, stream>>>(staticf, w2 + L * 2 * H, bias2 + L * H,
                                               vv[L], t, hy);
    }

    // relu(hy @ fc1_w^T + fc1_b) -> t (reuse)
    critic_gemm256_wmma<<<32, 256, 0, stream>>>(hy, fc1_w, fc1_b, t, 1);
    // out[b] = dot(t[b,:], fc2_w) + fc2_b
    critic_fc2<<<BATCH, H, 0, stream>>>(t, fc2_w, fc2_b, (float*)d_out);
}